// Backflow_88519275970596
// MI455X (gfx1250) — compile-verified
//
#include <hip/hip_runtime.h>

typedef __attribute__((ext_vector_type(2))) float v2f;
typedef __attribute__((ext_vector_type(8))) float v8f;

#define N_PART 1024
#define DIM 3
#define TILE 16
#define NJT (N_PART / TILE)       // 64 j-tiles
#define WAVES_PER_BLOCK 8
#define THREADS (WAVES_PER_BLOCK * 32)

// sum across the two 16-lane halves: v += swizzle_xor16(v)
__device__ __forceinline__ float half_sum(float v) {
    int i = __builtin_bit_cast(int, v);
    // ds_swizzle group-of-32: and=0x1f, or=0, xor=0x10 -> SWAPX16
    int s = __builtin_amdgcn_ds_swizzle(i, 0x401F);
    return v + __builtin_bit_cast(float, s);
}

__global__ void __launch_bounds__(THREADS)
Backflow_88519275970596_kernel(const float* __restrict__ x,
                               float* __restrict__ out) {
    __shared__ float4 tile[N_PART];   // {x, y, z, |x|^2} : 16 KB LDS

    const int tid  = threadIdx.x;
    const int lane = tid & 31;
    const int wave = tid >> 5;
    const int col  = lane & 15;
    const bool lo  = lane < 16;

    const int batch = blockIdx.x >> 3;   // 8 blocks per batch
    const int iblk  = blockIdx.x & 7;

    const float* xb = x + batch * (N_PART * DIM);

    // Stage the batch's particles in LDS with precomputed squared norms.
    for (int r = tid; r < N_PART; r += THREADS) {
        float a = xb[3 * r + 0];
        float b = xb[3 * r + 1];
        float c = xb[3 * r + 2];
        tile[r] = make_float4(a, b, c, a * a + b * b + c * c);
    }
    __syncthreads();

    // This wave owns i-rows [i0, i0+16).
    const int i0 = (iblk * WAVES_PER_BLOCK + wave) * TILE;
    const float4 xi = tile[i0 + col];
    const float qi  = xi.w;

    // B matrix (4x16, K=dim padded to 4): vgpr0 = K0 row (lo) / K2 row (hi),
    // vgpr1 = K1 row (lo) / K3=0 row (hi).
    v2f bmat;
    bmat.x = lo ? xi.x : xi.z;
    bmat.y = lo ? xi.y : 0.0f;

    const int hofs = lo ? 0 : 8;   // D layout: vgpr v holds M=v (lo) / v+8 (hi)

    float accS = 0.0f, accTx = 0.0f, accTy = 0.0f, accTz = 0.0f;

    const float KE = -0.48089834696298783f;  // -log2(e)/3 : eta = 0.8*exp2(KE*d)

    for (int jt = 0; jt < NJT; ++jt) {
        const int j0 = jt * TILE;

        // A matrix (16x4 rows of x_j): lane l<16 gets comps (x,y); l>=16 (z,0).
        const float4 xa = tile[j0 + col];
        v2f amat;
        amat.x = lo ? xa.x : xa.z;
        amat.y = lo ? xa.y : 0.0f;

        v8f zero = {};
        // G[M=j, N=i] = x_j . x_i  (Gram tile on the matrix pipe)
        v8f G = __builtin_amdgcn_wmma_f32_16x16x4_f32(
            false, amat, false, bmat, (short)0, zero, false, false);

        #pragma unroll
        for (int v = 0; v < 8; ++v) {
            // broadcast read: uniform address within each half-wave
            const float4 xj = tile[j0 + hofs + v];
            float d2 = fmaf(-2.0f, G[v], qi + xj.w);
            d2 = fmaxf(d2, 0.0f);               // guard fp-negative on diagonal
            const float d = __builtin_amdgcn_sqrtf(d2);
            const float e = __builtin_amdgcn_exp2f(d * KE);  // eta without 0.8
            accS  += e;
            accTx = fmaf(e, xj.x, accTx);
            accTy = fmaf(e, xj.y, accTy);
            accTz = fmaf(e, xj.z, accTz);
        }
    }

    // Combine the two 16-lane halves (M=0..7 partials + M=8..15 partials).
    accS  = half_sum(accS);
    accTx = half_sum(accTx);
    accTy = half_sum(accTy);
    accTz = half_sum(accTz);

    // out_i = 0.8 * (S_i * x_i - T_i); diagonal term cancels exactly.
    if (lo) {
        const int o = (batch * N_PART + i0 + col) * DIM;
        out[o + 0] = 0.8f * (accS * xi.x - accTx);
        out[o + 1] = 0.8f * (accS * xi.y - accTy);
        out[o + 2] = 0.8f * (accS * xi.z - accTz);
    }
}

extern "C" void kernel_launch(void* const* d_in, const int* in_sizes, int n_in,
                              void* d_out, int out_size, void* d_ws, size_t ws_size,
                              hipStream_t stream) {
    (void)n_in; (void)d_ws; (void)ws_size; (void)out_size;
    const float* x = (const float*)d_in[0];
    float* out = (float*)d_out;

    const int B = in_sizes[0] / (N_PART * DIM);   // 64
    const int blocks = B * 8;                      // 8 i-blocks per batch
    Backflow_88519275970596_kernel<<<blocks, THREADS, 0, stream>>>(x, out);
}